// EGCL_89764816487047
// MI455X (gfx1250) — compile-verified
//
#include <hip/hip_runtime.h>
#include <hip/hip_bf16.h>

// EGNN (EGCL) layer fused for gfx1250 (MI455X).
// Fusion: edge-MLP -> scatter -> node-MLP; the 413MB edge_in / 205MB
// edge_feat intermediates never touch HBM (node_feat = 12.8MB lives in the
// 192MB L2; segment sums are f32 atomics resolved in L2). All GEMMs use
// V_WMMA_F32_16X16X4_F32 (full f32 precision, matches the reference).
// Edge kernel processes 32 edges per wave (two 16-row A tiles) so every
// B-operand load feeds two WMMAs.

typedef __attribute__((ext_vector_type(2))) float v2f;
typedef __attribute__((ext_vector_type(8))) float v8f;

#define WAVES 4
#define TPB   (WAVES * 32)

__device__ __forceinline__ v8f wmma_f32(v2f a, v2f b, v8f c) {
  // (neg_a, A, neg_b, B, c_mod, C, reuse_a, reuse_b)
  return __builtin_amdgcn_wmma_f32_16x16x4_f32(false, a, false, b, (short)0, c,
                                               false, false);
}

__device__ __forceinline__ void lds_wait() {
  asm volatile("s_wait_dscnt 0" ::: "memory");
}

__device__ __forceinline__ v2f ld2(const float* p) {
  v2f a;
  a.x = p[0];
  a.y = p[1];
  return a;
}

// ---------------------------------------------------------------------------
// Workspace zero fill (graph-capture safe; harness poisons ws with 0xAA).
// ---------------------------------------------------------------------------
__global__ void egcl_zero_kernel(float* __restrict__ p, long n) {
  long i = (long)blockIdx.x * blockDim.x + threadIdx.x;
  long stride = (long)gridDim.x * blockDim.x;
  for (; i < n; i += stride) p[i] = 0.0f;
}

// ---------------------------------------------------------------------------
// Edge kernel: one wave == 32 edges (two 16-row M tiles sharing B operands).
//   GEMM1: edge_in[32x129(pad132)] @ We1[129x64]
//   GEMM2: h @ We2[64x64]            -> edge_feat (scattered to agg_edge)
//   GEMM3: edge_feat @ Wc1[64x64]    -> relu -> dot wc -> coord scatter
// A layout (16x4 f32): lanes 0-15 hold (m=lane, k, k+1); lanes 16-31 hold
// (m=lane-16, k+2, k+3). B mirrored (k rows split across halves).
// ---------------------------------------------------------------------------
__global__ void __launch_bounds__(TPB)
egcl_edge_kernel(const float* __restrict__ node_feat,
                 const float* __restrict__ coord,
                 const int*   __restrict__ edge_list,
                 const float* __restrict__ We1, const float* __restrict__ be1,
                 const float* __restrict__ We2, const float* __restrict__ be2,
                 const float* __restrict__ Wc1, const float* __restrict__ bc1,
                 const float* __restrict__ wc,
                 float* __restrict__ agg_edge,  // [N*64]
                 float* __restrict__ agg,       // [N*3]
                 float* __restrict__ cnt,       // [N]
                 int n_tiles) {
  __shared__ float stage_all[WAVES * 2 * 16 * 64];

  const int lane = threadIdx.x & 31;
  const int wave = threadIdx.x >> 5;
  const int tile = blockIdx.x * WAVES + wave;
  if (tile >= n_tiles) return;  // wave-uniform: EXEC all-ones for every WMMA

  float* st0 = stage_all + wave * (2 * 16 * 64);
  float* st1 = st0 + 16 * 64;

  const int e0   = tile * 32;
  const int m    = lane & 15;   // row within each 16-row tile owned by lane
  const int half = lane >> 4;   // 0: lanes 0-15, 1: lanes 16-31
  const int kk   = half << 1;   // A/B sub-k offset: 0 or 2
  const int col  = lane & 15;   // B/C column within a 16-wide n-tile

  // Edge endpoints for this lane's rows in both tiles.
  const int2 eA = ((const int2*)edge_list)[e0 + m];
  const int2 eB = ((const int2*)edge_list)[e0 + 16 + m];
  const int nin0 = eA.x, nout0 = eA.y;
  const int nin1 = eB.x, nout1 = eB.y;

  // coord_diff + radial for both rows (valid in every lane).
  const float cdx0 = coord[nout0 * 3 + 0] - coord[nin0 * 3 + 0];
  const float cdy0 = coord[nout0 * 3 + 1] - coord[nin0 * 3 + 1];
  const float cdz0 = coord[nout0 * 3 + 2] - coord[nin0 * 3 + 2];
  const float rad0 = cdx0 * cdx0 + cdy0 * cdy0 + cdz0 * cdz0;
  const float cdx1 = coord[nout1 * 3 + 0] - coord[nin1 * 3 + 0];
  const float cdy1 = coord[nout1 * 3 + 1] - coord[nin1 * 3 + 1];
  const float cdz1 = coord[nout1 * 3 + 2] - coord[nin1 * 3 + 2];
  const float rad1 = cdx1 * cdx1 + cdy1 * cdy1 + cdz1 * cdz1;

  const float* rin0  = node_feat + (size_t)nin0 * 64 + kk;
  const float* rout0 = node_feat + (size_t)nout0 * 64 + kk;
  const float* rin1  = node_feat + (size_t)nin1 * 64 + kk;
  const float* rout1 = node_feat + (size_t)nout1 * 64 + kk;

  // ---------------- GEMM1: (32x132) @ (132x64) -> h -----------------------
  v8f acc0[4] = {};
  v8f acc1[4] = {};
  for (int kc = 0; kc < 16; ++kc) {  // k 0..63: node_feat[node_in]
    const v2f a0 = ld2(rin0 + kc * 4);
    const v2f a1 = ld2(rin1 + kc * 4);
    const float* bp = We1 + (size_t)(kc * 4 + kk) * 64 + col;
#pragma unroll
    for (int nt = 0; nt < 4; ++nt) {
      v2f b;
      b.x = bp[nt * 16];
      b.y = bp[64 + nt * 16];
      acc0[nt] = wmma_f32(a0, b, acc0[nt]);
      acc1[nt] = wmma_f32(a1, b, acc1[nt]);
    }
  }
  for (int kc = 0; kc < 16; ++kc) {  // k 64..127: node_feat[node_out]
    const v2f a0 = ld2(rout0 + kc * 4);
    const v2f a1 = ld2(rout1 + kc * 4);
    const float* bp = We1 + (size_t)(64 + kc * 4 + kk) * 64 + col;
#pragma unroll
    for (int nt = 0; nt < 4; ++nt) {
      v2f b;
      b.x = bp[nt * 16];
      b.y = bp[64 + nt * 16];
      acc0[nt] = wmma_f32(a0, b, acc0[nt]);
      acc1[nt] = wmma_f32(a1, b, acc1[nt]);
    }
  }
  {  // k chunk 32: k=128 is radial, 129..131 zero-padded.
    // B row 128 is loaded by ALL lanes (in-bounds); half-1 lanes carry the
    // k=130/131 positions whose A entries are forced to zero, so the B
    // content there is don't-care. No divergent loads.
    v2f a0r, a1r;
    a0r.x = (half == 0) ? rad0 : 0.0f;
    a0r.y = 0.0f;
    a1r.x = (half == 0) ? rad1 : 0.0f;
    a1r.y = 0.0f;
    const float* bp = We1 + (size_t)128 * 64 + col;
#pragma unroll
    for (int nt = 0; nt < 4; ++nt) {
      v2f b;
      b.x = bp[nt * 16];
      b.y = 0.0f;
      acc0[nt] = wmma_f32(a0r, b, acc0[nt]);
      acc1[nt] = wmma_f32(a1r, b, acc1[nt]);
    }
  }
  // bias + relu, stage h row-major in LDS for A-layout reload.
#pragma unroll
  for (int nt = 0; nt < 4; ++nt) {
    const float bb = be1[nt * 16 + col];
#pragma unroll
    for (int r = 0; r < 8; ++r) {
      const int off = (half * 8 + r) * 64 + nt * 16 + col;
      st0[off] = fmaxf(acc0[nt][r] + bb, 0.0f);
      st1[off] = fmaxf(acc1[nt][r] + bb, 0.0f);
    }
  }
  lds_wait();

  // ---------------- GEMM2: h @ We2 -> edge_feat ----------------------------
  v8f ef0[4] = {};
  v8f ef1[4] = {};
  for (int kc = 0; kc < 16; ++kc) {
    const v2f a0 = ld2(st0 + m * 64 + kc * 4 + kk);
    const v2f a1 = ld2(st1 + m * 64 + kc * 4 + kk);
    const float* bp = We2 + (size_t)(kc * 4 + kk) * 64 + col;
#pragma unroll
    for (int nt = 0; nt < 4; ++nt) {
      v2f b;
      b.x = bp[nt * 16];
      b.y = bp[64 + nt * 16];
      ef0[nt] = wmma_f32(a0, b, ef0[nt]);
      ef1[nt] = wmma_f32(a1, b, ef1[nt]);
    }
  }
  // bias + relu; stage for GEMM3's A operand.
#pragma unroll
  for (int nt = 0; nt < 4; ++nt) {
    const float bb = be2[nt * 16 + col];
#pragma unroll
    for (int r = 0; r < 8; ++r) {
      const int off = (half * 8 + r) * 64 + nt * 16 + col;
      const float v0 = fmaxf(ef0[nt][r] + bb, 0.0f);
      const float v1 = fmaxf(ef1[nt][r] + bb, 0.0f);
      ef0[nt][r] = v0;
      ef1[nt][r] = v1;
      st0[off] = v0;
      st1[off] = v1;
    }
  }
  // Scatter agg_edge NOW so ef registers die before GEMM3's accumulators
  // come alive (GEMM3 reads the staged LDS copy).
#pragma unroll
  for (int r = 0; r < 8; ++r) {
    const int row = half * 8 + r;
    const int no0 = edge_list[2 * (e0 + row) + 1];
    const int no1 = edge_list[2 * (e0 + 16 + row) + 1];
    float* d0 = agg_edge + (size_t)no0 * 64 + col;
    float* d1 = agg_edge + (size_t)no1 * 64 + col;
    atomicAdd(d0 + 0, ef0[0][r]);
    atomicAdd(d0 + 16, ef0[1][r]);
    atomicAdd(d0 + 32, ef0[2][r]);
    atomicAdd(d0 + 48, ef0[3][r]);
    atomicAdd(d1 + 0, ef1[0][r]);
    atomicAdd(d1 + 16, ef1[1][r]);
    atomicAdd(d1 + 32, ef1[2][r]);
    atomicAdd(d1 + 48, ef1[3][r]);
  }
  lds_wait();

  // ---------------- GEMM3: edge_feat @ Wc1 -> relu -> dot wc ---------------
  v8f c30[4] = {};
  v8f c31[4] = {};
  for (int kc = 0; kc < 16; ++kc) {
    const v2f a0 = ld2(st0 + m * 64 + kc * 4 + kk);
    const v2f a1 = ld2(st1 + m * 64 + kc * 4 + kk);
    const float* bp = Wc1 + (size_t)(kc * 4 + kk) * 64 + col;
#pragma unroll
    for (int nt = 0; nt < 4; ++nt) {
      v2f b;
      b.x = bp[nt * 16];
      b.y = bp[64 + nt * 16];
      c30[nt] = wmma_f32(a0, b, c30[nt]);
      c31[nt] = wmma_f32(a1, b, c31[nt]);
    }
  }
  const float wv0 = wc[col];
  const float wv1 = wc[16 + col];
  const float wv2 = wc[32 + col];
  const float wv3 = wc[48 + col];
  const float bb0 = bc1[col];
  const float bb1 = bc1[16 + col];
  const float bb2 = bc1[32 + col];
  const float bb3 = bc1[48 + col];
  float p0[8], p1[8];
#pragma unroll
  for (int r = 0; r < 8; ++r) {
    p0[r] = fmaxf(c30[0][r] + bb0, 0.0f) * wv0 +
            fmaxf(c30[1][r] + bb1, 0.0f) * wv1 +
            fmaxf(c30[2][r] + bb2, 0.0f) * wv2 +
            fmaxf(c30[3][r] + bb3, 0.0f) * wv3;
    p1[r] = fmaxf(c31[0][r] + bb0, 0.0f) * wv0 +
            fmaxf(c31[1][r] + bb1, 0.0f) * wv1 +
            fmaxf(c31[2][r] + bb2, 0.0f) * wv2 +
            fmaxf(c31[3][r] + bb3, 0.0f) * wv3;
  }
#pragma unroll
  for (int mask = 1; mask < 16; mask <<= 1) {
#pragma unroll
    for (int r = 0; r < 8; ++r) {
      p0[r] += __shfl_xor(p0[r], mask, 32);
      p1[r] += __shfl_xor(p1[r], mask, 32);
    }
  }
  if (col == 0) {  // park trans_w: tile0 -> st0[0..15], tile1 -> st0[16..31]
#pragma unroll
    for (int r = 0; r < 8; ++r) {
      st0[half * 8 + r] = p0[r];
      st0[16 + half * 8 + r] = p1[r];
    }
  }
  lds_wait();

  // coord scatter (segment sums via L2 f32 atomics)
  if (half == 0) {  // lanes 0-15 own edges e0+m and e0+16+m
    const float tw0 = st0[m];
    const float tw1 = st0[16 + m];
    atomicAdd(&agg[nout0 * 3 + 0], cdx0 * tw0);
    atomicAdd(&agg[nout0 * 3 + 1], cdy0 * tw0);
    atomicAdd(&agg[nout0 * 3 + 2], cdz0 * tw0);
    atomicAdd(&cnt[nout0], 1.0f);
    atomicAdd(&agg[nout1 * 3 + 0], cdx1 * tw1);
    atomicAdd(&agg[nout1 * 3 + 1], cdy1 * tw1);
    atomicAdd(&agg[nout1 * 3 + 2], cdz1 * tw1);
    atomicAdd(&cnt[nout1], 1.0f);
  }
}

// ---------------------------------------------------------------------------
// Node kernel: one wave == 16 nodes.
//   GEMM1: [node_feat | agg_edge] (16x128) @ Wn1 (128x64), relu
//   GEMM2: (16x64) @ Wn2 (64x64), + bn2, + residual; coord update.
// ---------------------------------------------------------------------------
__global__ void __launch_bounds__(TPB)
egcl_node_kernel(const float* __restrict__ node_feat,
                 const float* __restrict__ coord,
                 const float* __restrict__ Wn1, const float* __restrict__ bn1,
                 const float* __restrict__ Wn2, const float* __restrict__ bn2,
                 const float* __restrict__ agg_edge,
                 const float* __restrict__ agg,
                 const float* __restrict__ cnt,
                 float* __restrict__ out_feat,   // [N*64]
                 float* __restrict__ out_coord,  // [N*3]
                 int n_tiles) {
  __shared__ float stage_all[WAVES * 16 * 64];

  const int lane = threadIdx.x & 31;
  const int wave = threadIdx.x >> 5;
  const int tile = blockIdx.x * WAVES + wave;
  if (tile >= n_tiles) return;

  float* stage = stage_all + wave * (16 * 64);

  const int m    = lane & 15;
  const int half = lane >> 4;
  const int kk   = half << 1;
  const int col  = lane & 15;

  const int nd = tile * 16 + m;
  const float* rowf = node_feat + (size_t)nd * 64 + kk;
  const float* rowa = agg_edge + (size_t)nd * 64 + kk;

  // GEMM1: (16x128) @ Wn1 -- split loops avoid per-iteration pointer selects
  v8f acc[4] = {};
  for (int kc = 0; kc < 16; ++kc) {
    const v2f a = ld2(rowf + kc * 4);
    const float* bp = Wn1 + (size_t)(kc * 4 + kk) * 64 + col;
#pragma unroll
    for (int nt = 0; nt < 4; ++nt) {
      v2f b;
      b.x = bp[nt * 16];
      b.y = bp[64 + nt * 16];
      acc[nt] = wmma_f32(a, b, acc[nt]);
    }
  }
  for (int kc = 0; kc < 16; ++kc) {
    const v2f a = ld2(rowa + kc * 4);
    const float* bp = Wn1 + (size_t)(64 + kc * 4 + kk) * 64 + col;
#pragma unroll
    for (int nt = 0; nt < 4; ++nt) {
      v2f b;
      b.x = bp[nt * 16];
      b.y = bp[64 + nt * 16];
      acc[nt] = wmma_f32(a, b, acc[nt]);
    }
  }
#pragma unroll
  for (int nt = 0; nt < 4; ++nt) {
    const float bb = bn1[nt * 16 + col];
#pragma unroll
    for (int r = 0; r < 8; ++r) {
      stage[(half * 8 + r) * 64 + nt * 16 + col] = fmaxf(acc[nt][r] + bb, 0.0f);
    }
  }
  lds_wait();

  // GEMM2: (16x64) @ Wn2
  v8f acc2[4] = {};
  for (int kc = 0; kc < 16; ++kc) {
    const v2f a = ld2(stage + m * 64 + kc * 4 + kk);
    const float* bp = Wn2 + (size_t)(kc * 4 + kk) * 64 + col;
#pragma unroll
    for (int nt = 0; nt < 4; ++nt) {
      v2f b;
      b.x = bp[nt * 16];
      b.y = bp[64 + nt * 16];
      acc2[nt] = wmma_f32(a, b, acc2[nt]);
    }
  }

  // residual write-out: out_feat = node_feat + (nf + bn2)
#pragma unroll
  for (int nt = 0; nt < 4; ++nt) {
    const float bb = bn2[nt * 16 + col];
#pragma unroll
    for (int r = 0; r < 8; ++r) {
      const int nd2 = tile * 16 + half * 8 + r;
      const size_t idx = (size_t)nd2 * 64 + nt * 16 + col;
      out_feat[idx] = node_feat[idx] + acc2[nt][r] + bb;
    }
  }

  // coord update (lanes 0-15, one node each)
  if (half == 0) {
    const float inv = 1.0f / fmaxf(cnt[nd], 1.0f);
    out_coord[nd * 3 + 0] = coord[nd * 3 + 0] + agg[nd * 3 + 0] * inv;
    out_coord[nd * 3 + 1] = coord[nd * 3 + 1] + agg[nd * 3 + 1] * inv;
    out_coord[nd * 3 + 2] = coord[nd * 3 + 2] + agg[nd * 3 + 2] * inv;
  }
}

// ---------------------------------------------------------------------------
extern "C" void kernel_launch(void* const* d_in, const int* in_sizes, int n_in,
                              void* d_out, int out_size, void* d_ws,
                              size_t ws_size, hipStream_t stream) {
  const float* node_feat = (const float*)d_in[0];
  const float* coord     = (const float*)d_in[1];
  const int*   edge_list = (const int*)d_in[2];
  const float* We1 = (const float*)d_in[3];
  const float* be1 = (const float*)d_in[4];
  const float* We2 = (const float*)d_in[5];
  const float* be2 = (const float*)d_in[6];
  const float* Wn1 = (const float*)d_in[7];
  const float* bn1 = (const float*)d_in[8];
  const float* Wn2 = (const float*)d_in[9];
  const float* bn2 = (const float*)d_in[10];
  const float* Wc1 = (const float*)d_in[11];
  const float* bc1 = (const float*)d_in[12];
  const float* wc  = (const float*)d_in[13];

  const int N = in_sizes[0] / 64;  // 50000
  const int E = in_sizes[2] / 2;   // 800000 (multiple of 32)

  float* agg_edge = (float*)d_ws;                   // N*64
  float* agg      = agg_edge + (size_t)N * 64;      // N*3
  float* cnt      = agg + (size_t)N * 3;            // N
  float* out_feat  = (float*)d_out;                 // N*64
  float* out_coord = out_feat + (size_t)N * 64;     // N*3

  // 1) zero the accumulators
  const long zn = (long)N * 68;
  egcl_zero_kernel<<<512, 256, 0, stream>>>(agg_edge, zn);

  // 2) fused edge MLP + scatter (32 edges per wave)
  const int etiles  = E / 32;
  const int eblocks = (etiles + WAVES - 1) / WAVES;
  egcl_edge_kernel<<<eblocks, TPB, 0, stream>>>(
      node_feat, coord, edge_list, We1, be1, We2, be2, Wc1, bc1, wc, agg_edge,
      agg, cnt, etiles);

  // 3) node MLP + residual + coord update (16 nodes per wave)
  const int ntiles  = N / 16;
  const int nblocks = (ntiles + WAVES - 1) / WAVES;
  egcl_node_kernel<<<nblocks, TPB, 0, stream>>>(
      node_feat, coord, Wn1, bn1, Wn2, bn2, agg_edge, agg, cnt, out_feat,
      out_coord, ntiles);
}